// SyntheticTripletLoss_9861244912210
// MI455X (gfx1250) — compile-verified
//
#include <hip/hip_runtime.h>

// ---------------------------------------------------------------------------
// SyntheticTripletLoss on MI455X (gfx1250, wave32)
//
// Closed form per row r=(b,t):  pp=p.p, pt=p.t, tt=t.t
//   sim_pn = (pp - pt^2) / max(sqrt(pp - 2 pt^2 + pt^2 tt), eps)
//   loss   = max(0.5 + sim_pn - pt, 0), masked mean over t < len[b]
//
// Memory-bound (64 MiB stream, ~2.9us floor @ 23.3 TB/s). One wave handles a
// 16-row tile; pt comes from the diagonal of V_WMMA_F32_16X16X4_F32 Gram
// tiles (A-fragment layout == row-major b64 loads per lane), pp/tt from
// per-lane FMAs on the same registers. Deterministic 2-pass reduction.
// ---------------------------------------------------------------------------

typedef __attribute__((ext_vector_type(2))) float v2f;
typedef __attribute__((ext_vector_type(8))) float v8f;

namespace {
constexpr int   kB = 32;
constexpr int   kT = 512;
constexpr int   kD = 512;
constexpr int   kRows = kB * kT;                 // 16384
constexpr int   kTileRows = 16;
constexpr int   kTiles = kRows / kTileRows;      // 1024
constexpr int   kWavesPerBlock = 8;              // 256 threads
constexpr int   kBlocks = kTiles / kWavesPerBlock;
constexpr float kMargin = 0.5f;
constexpr float kEps = 1e-12f;
}

__global__ __launch_bounds__(256)
void triplet_tile_kernel(const float* __restrict__ preds,
                         const float* __restrict__ targets,
                         const int*   __restrict__ lens,
                         float*       __restrict__ tileSums) {
  const int lane = threadIdx.x & 31;
  const int wave = threadIdx.x >> 5;
  const int tile = blockIdx.x * kWavesPerBlock + wave;   // 0..kTiles-1
  const int rowBase = tile * kTileRows;

  // WMMA f32 16x16x4 A-fragment addressing:
  //   lanes 0..15  hold row m = lane,     K = {k0,   k0+1}
  //   lanes 16..31 hold row m = lane-16,  K = {k0+2, k0+3}
  const int m  = lane & 15;
  const int kh = lane >> 4;
  const float* pA = preds   + (size_t)(rowBase + m) * kD + 2 * kh;
  const float* tA = targets + (size_t)(rowBase + m) * kD + 2 * kh;

  v8f  accPT = {};          // Gram(P, T^T) accumulator; diag = pt per row
  float pp = 0.0f, tt = 0.0f;

#pragma unroll 4
  for (int k0 = 0; k0 < kD; k0 += 4) {
    v2f a = *(const v2f*)(pA + k0);   // global_load_b64 (8B aligned)
    v2f b = *(const v2f*)(tA + k0);
    // per-lane partial pp/tt (row m, this lane's K slice)
    pp = __builtin_fmaf(a.x, a.x, pp);
    pp = __builtin_fmaf(a.y, a.y, pp);
    tt = __builtin_fmaf(b.x, b.x, tt);
    tt = __builtin_fmaf(b.y, b.y, tt);
    // D += A(16x4) x B(4x16); B fragment of T^T == A-style load of T
    accPT = __builtin_amdgcn_wmma_f32_16x16x4_f32(
        /*neg_a=*/false, a, /*neg_b=*/false, b,
        /*c_mod=*/(short)0, accPT, /*reuse_a=*/false, /*reuse_b=*/false);
  }

  // ---- extract diag(C): C VGPR j holds M=j (lanes 0-15) / M=j+8 (16-31) ----
  float dsel = 0.0f;
#pragma unroll
  for (int j = 0; j < 8; ++j) {
    float cj = accPT[j];
    if (lane == j)      dsel = cj;          // rows 0..7  live in lanes 0..7
    if (lane == j + 24) dsel = cj;          // rows 8..15 live in lanes 24..31
  }
  const int src = (m < 8) ? m : (m + 16);
  const float pt  = __shfl(dsel, src, 32);
  // combine the two half-wave K slices for pp/tt of row m
  const float ppr = __shfl(pp, m, 32) + __shfl(pp, m + 16, 32);
  const float ttr = __shfl(tt, m, 32) + __shfl(tt, m + 16, 32);

  // ---- closed-form loss for row m ----
  const float pt2   = pt * pt;
  const float pnum  = ppr - pt2;                                  // p.neg
  const float nn    = __builtin_fmaf(pt2, ttr, ppr) - 2.0f * pt2; // |neg|^2
  const float nrm   = __builtin_sqrtf(fmaxf(nn, 0.0f));
  const float simpn = pnum / fmaxf(nrm, kEps);
  const float loss  = fmaxf(kMargin + simpn - pt, 0.0f);

  const int row  = rowBase + m;
  const int bidx = row / kT;                 // uniform within the tile
  const int tidx = row - bidx * kT;
  float contrib = (lane < 16 && tidx < lens[bidx]) ? loss : 0.0f;

  // wave butterfly sum (deterministic)
#pragma unroll
  for (int off = 16; off > 0; off >>= 1)
    contrib += __shfl_xor(contrib, off, 32);
  if (lane == 0) tileSums[tile] = contrib;
}

__global__ __launch_bounds__(256)
void triplet_finalize_kernel(const float* __restrict__ tileSums,
                             const int*   __restrict__ lens,
                             float*       __restrict__ out) {
  __shared__ float sh[256];
  const int t = threadIdx.x;
  float s = 0.0f;
  for (int i = t; i < kTiles; i += 256) s += tileSums[i];  // fixed order
  sh[t] = s;
  __syncthreads();
#pragma unroll
  for (int off = 128; off > 0; off >>= 1) {
    if (t < off) sh[t] += sh[t + off];
    __syncthreads();
  }
  if (t == 0) {
    int v = 0;
    for (int b = 0; b < kB; ++b) v += lens[b];  // sum of valid tokens
    out[0] = sh[0] / (float)v;
  }
}

extern "C" void kernel_launch(void* const* d_in, const int* in_sizes, int n_in,
                              void* d_out, int out_size, void* d_ws, size_t ws_size,
                              hipStream_t stream) {
  const float* preds   = (const float*)d_in[0];
  const float* targets = (const float*)d_in[1];
  const int*   lens    = (const int*)d_in[2];
  float* tileSums = (float*)d_ws;           // kTiles floats, fully overwritten

  triplet_tile_kernel<<<kBlocks, 256, 0, stream>>>(preds, targets, lens, tileSums);
  triplet_finalize_kernel<<<1, 256, 0, stream>>>(tileSums, lens, (float*)d_out);
}